// TangentProjections_36086315221234
// MI455X (gfx1250) — compile-verified
//
#include <hip/hip_runtime.h>
#include <math.h>

// Problem constants (fixed by the reference: (4, 4096, 3) f32 in, (4, 4096, 25) f32 out)
#define NCLOUD 4
#define NPTS   4096
#define LIM    20      // NEIGHBOR_LIMIT
#define KNN    16      // N_NEIGHBORS (radius = mean of dsort[:,16])
#define NBINS  5
#define EPSV   1e-12f
#define NTILES (NPTS / 16)

typedef float v2f __attribute__((ext_vector_type(2)));
typedef float v8f __attribute__((ext_vector_type(8)));

// -------------------------------------------------------------------------
// Kernel 1: pack points as (x, y, z, ||p||^2) for single-b128 fragment loads,
// and zero the per-cloud radius accumulators.
// -------------------------------------------------------------------------
__global__ void tp_prep_kernel(const float* __restrict__ coords,
                               float4* __restrict__ pts4,
                               float* __restrict__ radsum) {
  int gid = blockIdx.x * blockDim.x + threadIdx.x;
  if (gid < NCLOUD) radsum[gid] = 0.0f;
  if (gid >= NCLOUD * NPTS) return;
  float x = coords[gid * 3 + 0];
  float y = coords[gid * 3 + 1];
  float z = coords[gid * 3 + 2];
  pts4[gid] = make_float4(x, y, z, x * x + y * y + z * z);
}

// -------------------------------------------------------------------------
// Kernel 2: WMMA distance tiles + per-row streaming top-20 selection.
//
// dist^2(i,j) = ||p_i||^2 - 2 p_i.p_j + ||p_j||^2 is computed as a single
// V_WMMA_F32_16X16X4_F32:  A = [x  y  z  ||p||^2] (16x4, row tile)
//                          B = [-2x -2y -2z 1]^T  (4x16, col tile)
//                          C = splat(||q_col||^2)
// Each wave owns 32 rows (two 16-row A fragments); each block is 2 waves
// (64 threads, wave32). The 16x16 D tile goes through LDS so that each of
// the 32 lanes owns one row and maintains a sorted top-20 in registers.
// -------------------------------------------------------------------------
__global__ __launch_bounds__(64) void tp_topk_kernel(
    const float4* __restrict__ pts4,
    int* __restrict__ nidx,
    float* __restrict__ ndist,
    float* __restrict__ radsum) {
  __shared__ float ldsd[2][32][17];  // [wave][local row][col], +1 pad vs banks

  const int wave = threadIdx.x >> 5;
  const int lane = threadIdx.x & 31;
  const int l16  = lane & 15;
  const int hi   = lane >> 4;                    // 0: K={0,1} half, 1: K={2,3} half
  const int b    = blockIdx.x >> 6;              // 64 blocks per cloud
  const int rowBase = ((blockIdx.x & 63) << 6) + (wave << 5);  // 64 rows/block, 32/wave
  const float4* __restrict__ P = pts4 + b * NPTS;

  // A fragments for the wave's two row tiles.
  // ISA layout (16x4 f32 A): lanes 0-15 hold M=lane, K=0..1; lanes 16-31 hold M=lane-16, K=2..3.
  float4 pr0 = P[rowBase + l16];
  float4 pr1 = P[rowBase + 16 + l16];
  v2f a0, a1;
  if (hi) { a0.x = pr0.z; a0.y = pr0.w; a1.x = pr1.z; a1.y = pr1.w; }
  else    { a0.x = pr0.x; a0.y = pr0.y; a1.x = pr1.x; a1.y = pr1.y; }

  // Sorted ascending top-20 (dist^2, idx) in registers.
  float key[LIM];
  int   kid[LIM];
#pragma unroll
  for (int j = 0; j < LIM; ++j) { key[j] = 3.4e38f; kid[j] = 0; }

  for (int ct = 0; ct < NTILES; ++ct) {
    float4 pc = P[ct * 16 + l16];
    __builtin_prefetch(&P[((ct + 1) & (NTILES - 1)) * 16 + l16], 0, 0);

    // B fragment (4x16): lanes 0-15 hold K=0..1, lanes 16-31 hold K=2..3, N = lane%16.
    v2f bf;
    if (hi) { bf.x = -2.0f * pc.z; bf.y = 1.0f; }
    else    { bf.x = -2.0f * pc.x; bf.y = -2.0f * pc.y; }

    // C fragment: every row of lane's column N gets ||q_N||^2.
    v8f cf;
#pragma unroll
    for (int r = 0; r < 8; ++r) cf[r] = pc.w;

    v8f d0 = __builtin_amdgcn_wmma_f32_16x16x4_f32(
        false, a0, false, bf, (short)0, cf, false, false);
    v8f d1 = __builtin_amdgcn_wmma_f32_16x16x4_f32(
        false, a1, false, bf, (short)0, cf, false, false);

    // D layout: VGPR r holds row (r + 8*hi), column l16.
#pragma unroll
    for (int r = 0; r < 8; ++r) {
      ldsd[wave][r + 8 * hi][l16]      = d0[r];
      ldsd[wave][16 + r + 8 * hi][l16] = d1[r];
    }
    __syncthreads();

    // Lane owns local row == lane; scan the 16 columns of this tile.
#pragma unroll
    for (int j = 0; j < 16; ++j) {
      float cd = fabsf(ldsd[wave][lane][j]);  // reference: sqrt(abs(.)), abs keeps order
      int   ci = ct * 16 + j;
      if (cd < key[LIM - 1]) {
#pragma unroll
        for (int t = 0; t < LIM; ++t) {
          bool sw = cd < key[t];
          float tk = key[t]; int ti = kid[t];
          key[t] = sw ? cd : tk;  kid[t] = sw ? ci : ti;
          cd     = sw ? tk : cd;  ci     = sw ? ti : ci;
        }
      }
    }
    __syncthreads();
  }

  const int row = rowBase + lane;
  atomicAdd(&radsum[b], sqrtf(key[KNN]));  // dsort[:,16] contribution

  long base = (long)(b * NPTS + row) * LIM;
#pragma unroll
  for (int j = 0; j < LIM; ++j) {
    nidx[base + j]  = kid[j];
    ndist[base + j] = sqrtf(key[j]);
  }
}

// -------------------------------------------------------------------------
// Kernel 3: SHOT LRF (analytic 3x3 symmetric eigensolver) + log-map +
// weighted 5x5 histogram. One thread per point; everything in registers.
// -------------------------------------------------------------------------
__device__ inline void tp_eigvec(float m00, float a01, float a02,
                                 float m11, float a12, float m22,
                                 float& vx, float& vy, float& vz) {
  // rows of (A - lambda I): r0=(m00,a01,a02) r1=(a01,m11,a12) r2=(a02,a12,m22)
  float c0x = a01 * a12 - a02 * m11, c0y = a02 * a01 - m00 * a12, c0z = m00 * m11 - a01 * a01;
  float c1x = a01 * m22 - a02 * a12, c1y = a02 * a02 - m00 * m22, c1z = m00 * a12 - a01 * a02;
  float c2x = m11 * m22 - a12 * a12, c2y = a12 * a02 - a01 * m22, c2z = a01 * a12 - m11 * a02;
  float n0 = c0x * c0x + c0y * c0y + c0z * c0z;
  float n1 = c1x * c1x + c1y * c1y + c1z * c1z;
  float n2 = c2x * c2x + c2y * c2y + c2z * c2z;
  float bx = c0x, by = c0y, bz = c0z, bn = n0;
  if (n1 > bn) { bx = c1x; by = c1y; bz = c1z; bn = n1; }
  if (n2 > bn) { bx = c2x; by = c2y; bz = c2z; bn = n2; }
  float s = rsqrtf(bn + 1e-30f);
  vx = bx * s; vy = by * s; vz = bz * s;
}

__global__ __launch_bounds__(128) void tp_shot_kernel(
    const float4* __restrict__ pts4,
    const int* __restrict__ nidx,
    const float* __restrict__ ndist,
    const float* __restrict__ radsum,
    float* __restrict__ out) {
  int gid = blockIdx.x * blockDim.x + threadIdx.x;
  if (gid >= NCLOUD * NPTS) return;
  int b = gid / NPTS;
  float radius = radsum[b] * (1.0f / NPTS);
  float fill = 2.0f * radius * 0.5773502691896258f;  // 2r/sqrt(3)
  float4 pi = pts4[gid];
  const float4* __restrict__ P = pts4 + b * NPTS;
  long base = (long)gid * LIM;

  float nx[LIM], ny[LIM], nz[LIM], nd[LIM];
#pragma unroll
  for (int k = 0; k < LIM; ++k) {
    int j = nidx[base + k];
    float dk = ndist[base + k];
    float4 q = P[j];
    float ax = q.x - pi.x, ay = q.y - pi.y, az = q.z - pi.z;
    if (dk > radius) { ax = fill; ay = fill; az = fill; }
    nx[k] = ax; ny[k] = ay; nz[k] = az;
    nd[k] = sqrtf(ax * ax + ay * ay + az * az);
  }

  // weighted covariance
  float wsum = 0.0f, c00 = 0, c01 = 0, c02 = 0, c11 = 0, c12 = 0, c22 = 0;
#pragma unroll
  for (int k = 0; k < LIM; ++k) {
    float w = fmaxf(radius - nd[k], 0.0f);
    wsum += w;
    c00 += w * nx[k] * nx[k]; c01 += w * nx[k] * ny[k]; c02 += w * nx[k] * nz[k];
    c11 += w * ny[k] * ny[k]; c12 += w * ny[k] * nz[k]; c22 += w * nz[k] * nz[k];
  }
  float cinv = 1.0f / (wsum + EPSV);
  c00 *= cinv; c01 *= cinv; c02 *= cinv; c11 *= cinv; c12 *= cinv; c22 *= cinv;

  // analytic eigenvalues (trigonometric method), ascending order
  float q3 = (c00 + c11 + c22) * (1.0f / 3.0f);
  float p1 = c01 * c01 + c02 * c02 + c12 * c12;
  float d00 = c00 - q3, d11 = c11 - q3, d22 = c22 - q3;
  float p2 = d00 * d00 + d11 * d11 + d22 * d22 + 2.0f * p1;
  float p = sqrtf(p2 * (1.0f / 6.0f)) + 1e-30f;
  float ip = 1.0f / p;
  float b00 = d00 * ip, b01 = c01 * ip, b02 = c02 * ip;
  float b11 = d11 * ip, b12 = c12 * ip, b22 = d22 * ip;
  float detB = b00 * (b11 * b22 - b12 * b12)
             - b01 * (b01 * b22 - b12 * b02)
             + b02 * (b01 * b12 - b11 * b02);
  float rr  = fminf(fmaxf(0.5f * detB, -1.0f), 1.0f);
  float phi = acosf(rr) * (1.0f / 3.0f);
  float lmax = q3 + 2.0f * p * cosf(phi);                       // eigvecs[:,2] -> x axis
  float lmin = q3 + 2.0f * p * cosf(phi + 2.0943951023931953f); // eigvecs[:,0] -> z axis

  float xx, xy, xz, zx, zy, zz;
  tp_eigvec(c00 - lmax, c01, c02, c11 - lmax, c12, c22 - lmax, xx, xy, xz);
  tp_eigvec(c00 - lmin, c01, c02, c11 - lmin, c12, c22 - lmin, zx, zy, zz);

  // sign disambiguation: sign = (#dots>=0 >= #dots<0) ? +1 : -1
  int sx = 0, sz = 0;
#pragma unroll
  for (int k = 0; k < LIM; ++k) {
    float dx = nx[k] * xx + ny[k] * xy + nz[k] * xz;
    float dz = nx[k] * zx + ny[k] * zy + nz[k] * zz;
    sx += (dx >= 0.0f) ? 1 : -1;
    sz += (dz >= 0.0f) ? 1 : -1;
  }
  if (sx < 0) { xx = -xx; xy = -xy; xz = -xz; }
  if (sz < 0) { zx = -zx; zy = -zy; zz = -zz; }

  // y = cross(z, x)
  float yx = zy * xz - zz * xy;
  float yy = zz * xx - zx * xz;
  float yz = zx * xy - zy * xx;

  // tangent-plane log map + 5x5 histogram
  float hist[NBINS * NBINS];
#pragma unroll
  for (int f = 0; f < NBINS * NBINS; ++f) hist[f] = 0.0f;
  float step  = 2.0f * radius * (1.0f / NBINS);
  float istep = 1.0f / step;

#pragma unroll
  for (int k = 0; k < LIM; ++k) {
    float u = nx[k] * xx + ny[k] * xy + nz[k] * xz;
    float v = nx[k] * yx + ny[k] * yy + nz[k] * yz;
    float pn = sqrtf(u * u + v * v);
    float s = nd[k] / (pn + EPSV);
    float uu = u * s, vv = v * s;
    float bxf = floorf((uu + radius) * istep);
    float byf = floorf((vv + radius) * istep);
    bool valid = (bxf >= 0.0f) && (bxf < (float)NBINS) &&
                 (byf >= 0.0f) && (byf < (float)NBINS);
    int bx = (int)bxf; bx = bx < 0 ? 0 : (bx > NBINS - 1 ? NBINS - 1 : bx);
    int by = (int)byf; by = by < 0 ? 0 : (by > NBINS - 1 ? NBINS - 1 : by);
    int flat = by * NBINS + bx;
    float one = valid ? 1.0f : 0.0f;
#pragma unroll
    for (int f = 0; f < NBINS * NBINS; ++f) hist[f] += (f == flat) ? one : 0.0f;
  }

  // radial weights + L2 normalization (reference has no eps here)
  float nrm2 = 0.0f;
#pragma unroll
  for (int f = 0; f < NBINS * NBINS; ++f) {
    int by = f / NBINS, bx = f - by * NBINS;
    float cy = -radius + step * ((float)by + 0.5f);
    float cx = -radius + step * ((float)bx + 0.5f);
    float wgt = radius - sqrtf(cy * cy + cx * cx);
    float h = hist[f] * wgt;
    hist[f] = h;
    nrm2 += h * h;
  }
  float rinv = 1.0f / sqrtf(nrm2);
  long obase = (long)gid * (NBINS * NBINS);
#pragma unroll
  for (int f = 0; f < NBINS * NBINS; ++f) out[obase + f] = hist[f] * rinv;
}

// -------------------------------------------------------------------------
// Workspace layout (all within d_ws):
//   pts4   : 16384 x float4   (256 KiB)
//   radsum : 4 x float
//   nidx   : 16384 x 20 x int (1.25 MiB)
//   ndist  : 16384 x 20 x f32 (1.25 MiB)
// -------------------------------------------------------------------------
extern "C" void kernel_launch(void* const* d_in, const int* in_sizes, int n_in,
                              void* d_out, int out_size, void* d_ws, size_t ws_size,
                              hipStream_t stream) {
  const float* coords = (const float*)d_in[0];
  float* out = (float*)d_out;

  float4* pts4  = (float4*)d_ws;
  float* radsum = (float*)(pts4 + NCLOUD * NPTS);
  int*   nidx   = (int*)(radsum + 4);
  float* ndist  = (float*)(nidx + NCLOUD * NPTS * LIM);

  tp_prep_kernel<<<(NCLOUD * NPTS + 255) / 256, 256, 0, stream>>>(coords, pts4, radsum);
  tp_topk_kernel<<<(NCLOUD * NPTS) / 64, 64, 0, stream>>>(pts4, nidx, ndist, radsum);
  tp_shot_kernel<<<(NCLOUD * NPTS + 127) / 128, 128, 0, stream>>>(pts4, nidx, ndist, radsum, out);
}